// MultiHeadAttention_13245679141144
// MI455X (gfx1250) — compile-verified
//
#include <hip/hip_runtime.h>
#include <hip/hip_bf16.h>

typedef __attribute__((ext_vector_type(2))) float v2f;
typedef __attribute__((ext_vector_type(8))) float v8f;

#define BATCH 2
#define SLEN  2048
#define EMB   512
#define NH    8
#define DH    64
#define HWIN  128
#define QKVN  1536   // 3*EMB
#define HSTRIDE 192  // 3*DH per head inside qkv row

// ---------------------------------------------------------------------------
// Generic fp32 WMMA GEMM: C[M,N] = A[M,K] @ W[K,N] + bias[N]
// One wave -> 16(M) x 64(N) tile, K-steps of 4 via v_wmma_f32_16x16x4_f32.
// ---------------------------------------------------------------------------
__global__ void gemm_wmma(const float* __restrict__ A,
                          const float* __restrict__ W,
                          const float* __restrict__ bias,
                          float* __restrict__ C,
                          int M, int K, int N) {
    const int lane = threadIdx.x & 31;
    const int wave = threadIdx.x >> 5;
    const int lm   = lane & 15;   // M (A) / N (B,C/D) index within tile
    const int half = lane >> 4;   // K sub-chunk selector

    const int ntiles = N >> 6;                    // 64-wide N tiles
    const int mtiles = M >> 4;
    const int wavesPerBlock = blockDim.x >> 5;
    int tile = blockIdx.x * wavesPerBlock + wave;
    if (tile >= ntiles * mtiles) return;          // wave-uniform exit
    const int mt = tile / ntiles;
    const int nt = tile % ntiles;
    const int m0 = mt << 4;
    const int n0 = nt << 6;

    v8f acc0 = {}, acc1 = {}, acc2 = {}, acc3 = {};

    const float* arow  = A + (size_t)(m0 + lm) * K + half * 2;
    const float* wbase = W + (size_t)(half * 2) * N + n0 + lm;

#pragma unroll 4
    for (int k = 0; k < K; k += 4) {
        // A fragment: a[j] = A[m0+lm, k + half*2 + j]  (8B-aligned pair)
        v2f a = *(const v2f*)(arow + k);
        const float* w0 = wbase + (size_t)k * N;
        const float* w1 = w0 + N;
        v2f b0, b1, b2, b3;
        b0.x = w0[0];  b0.y = w1[0];
        b1.x = w0[16]; b1.y = w1[16];
        b2.x = w0[32]; b2.y = w1[32];
        b3.x = w0[48]; b3.y = w1[48];
        acc0 = __builtin_amdgcn_wmma_f32_16x16x4_f32(false, a, false, b0, (short)0, acc0, false, false);
        acc1 = __builtin_amdgcn_wmma_f32_16x16x4_f32(false, a, false, b1, (short)0, acc1, false, false);
        acc2 = __builtin_amdgcn_wmma_f32_16x16x4_f32(false, a, false, b2, (short)0, acc2, false, false);
        acc3 = __builtin_amdgcn_wmma_f32_16x16x4_f32(false, a, false, b3, (short)0, acc3, false, false);
    }

    const float bv0 = bias[n0 + lm];
    const float bv1 = bias[n0 + 16 + lm];
    const float bv2 = bias[n0 + 32 + lm];
    const float bv3 = bias[n0 + 48 + lm];
#pragma unroll
    for (int i = 0; i < 8; ++i) {
        const size_t row = (size_t)(m0 + i + 8 * half) * N + n0 + lm;
        C[row]      = acc0[i] + bv0;
        C[row + 16] = acc1[i] + bv1;
        C[row + 32] = acc2[i] + bv2;
        C[row + 48] = acc3[i] + bv3;
    }
}

// ---------------------------------------------------------------------------
// Banded flash attention. One wave per (b, h, 16-query tile).
// scores^T = K @ Q^T  (softmax axis in VGPR rows -> cheap reduction)
// O^T      = V^T @ P^T (P^T staged through per-wave LDS tile)
// ---------------------------------------------------------------------------
__global__ void attn_banded(const float* __restrict__ qkv,
                            const int*   __restrict__ pad,
                            float*       __restrict__ out /* [B*S, EMB] */) {
    __shared__ float sP[4][16][17];

    const int lane = threadIdx.x & 31;
    const int wave = threadIdx.x >> 5;
    const int lm   = lane & 15;
    const int half = lane >> 4;

    const int QT = SLEN / 16;                 // 128 query tiles per (b,h)
    int gw = blockIdx.x * 4 + wave;
    int qt = gw % QT;
    int bh = gw / QT;
    int h  = bh % NH;
    int b  = bh / NH;
    if (b >= BATCH) return;                   // wave-uniform

    const int t0 = qt << 4;
    const int qg = t0 + lm;                   // this lane's query index

    // Q fragments (B operand of scores^T), pre-scaled by 1/sqrt(DH)=0.125
    const float* qptr = qkv + ((size_t)(b * SLEN + qg)) * QKVN + h * HSTRIDE + half * 2;
    v2f qb[16];
#pragma unroll
    for (int dc = 0; dc < 16; ++dc) {
        v2f t = *(const v2f*)(qptr + dc * 4);
        qb[dc].x = t.x * 0.125f;
        qb[dc].y = t.y * 0.125f;
    }
    const int pm_q = pad[b * SLEN + qg];

    float m_run = -1e30f;
    float l_run = 0.0f;
    v8f acc0 = {}, acc1 = {}, acc2 = {}, acc3 = {};

    const int kt0 = (qt - 8 < 0) ? 0 : qt - 8;
    const int kt1 = (qt + 8 > QT - 1) ? QT - 1 : qt + 8;

    for (int kt = kt0; kt <= kt1; ++kt) {
        const int k0 = kt << 4;

        // ---- scores^T = K_tile @ Q^T : 16 WMMA over d ----
        v8f sc = {};
        const float* kptr = qkv + ((size_t)(b * SLEN + k0 + lm)) * QKVN
                          + h * HSTRIDE + DH + half * 2;
#pragma unroll
        for (int dc = 0; dc < 16; ++dc) {
            v2f a = *(const v2f*)(kptr + dc * 4);
            sc = __builtin_amdgcn_wmma_f32_16x16x4_f32(false, a, false, qb[dc], (short)0, sc, false, false);
        }

        // ---- band + padding mask, online softmax ----
        float p[8];
        float tmax = -1e30f;
#pragma unroll
        for (int i = 0; i < 8; ++i) {
            const int key = k0 + i + 8 * half;
            const int d   = key - qg;
            const bool valid = (d <= HWIN) && (d >= -HWIN) && (pm_q != 0)
                             && (pad[b * SLEN + key] != 0);
            const float s = valid ? sc[i] : -1e30f;
            p[i] = s;
            tmax = fmaxf(tmax, s);
        }
        tmax = fmaxf(tmax, __shfl_xor(tmax, 16, 32));
        const float m_new = fmaxf(m_run, tmax);
        float tsum = 0.0f;
#pragma unroll
        for (int i = 0; i < 8; ++i) {
            const float pe = (p[i] > -5e29f) ? __expf(p[i] - m_new) : 0.0f;
            p[i] = pe;
            tsum += pe;
        }
        tsum += __shfl_xor(tsum, 16, 32);
        const float alpha = __expf(m_run - m_new);
        l_run = l_run * alpha + tsum;
        m_run = m_new;
        acc0 = acc0 * alpha; acc1 = acc1 * alpha;
        acc2 = acc2 * alpha; acc3 = acc3 * alpha;

        // ---- stage P^T [key][query] in per-wave LDS tile ----
#pragma unroll
        for (int i = 0; i < 8; ++i)
            sP[wave][i + 8 * half][lm] = p[i];

        // ---- O^T += V^T @ P^T : 16 WMMA (4 d-tiles x 4 key-chunks) ----
        const float* vptr = qkv + ((size_t)(b * SLEN + k0)) * QKVN + h * HSTRIDE + 2 * DH;
#pragma unroll
        for (int kc = 0; kc < 16; kc += 4) {
            const int keyr = kc + half * 2;
            v2f bfrag;
            bfrag.x = sP[wave][keyr][lm];
            bfrag.y = sP[wave][keyr + 1][lm];
            const float* v0 = vptr + (size_t)keyr * QKVN + lm;
            const float* v1 = v0 + QKVN;
            v2f a;
            a.x = v0[0];  a.y = v1[0];
            acc0 = __builtin_amdgcn_wmma_f32_16x16x4_f32(false, a, false, bfrag, (short)0, acc0, false, false);
            a.x = v0[16]; a.y = v1[16];
            acc1 = __builtin_amdgcn_wmma_f32_16x16x4_f32(false, a, false, bfrag, (short)0, acc1, false, false);
            a.x = v0[32]; a.y = v1[32];
            acc2 = __builtin_amdgcn_wmma_f32_16x16x4_f32(false, a, false, bfrag, (short)0, acc2, false, false);
            a.x = v0[48]; a.y = v1[48];
            acc3 = __builtin_amdgcn_wmma_f32_16x16x4_f32(false, a, false, bfrag, (short)0, acc3, false, false);
        }
    }

    // ---- normalize and write: out[b*S + q, h*64 + d] ----
    const float inv = (l_run > 0.0f && pm_q != 0) ? (1.0f / l_run) : 0.0f;
    float* optr = out + ((size_t)(b * SLEN + qg)) * EMB + h * DH;
#pragma unroll
    for (int i = 0; i < 8; ++i) {
        const int dr = i + 8 * half;
        optr[dr]      = acc0[i] * inv;
        optr[16 + dr] = acc1[i] * inv;
        optr[32 + dr] = acc2[i] * inv;
        optr[48 + dr] = acc3[i] * inv;
    }
}

// ---------------------------------------------------------------------------
extern "C" void kernel_launch(void* const* d_in, const int* in_sizes, int n_in,
                              void* d_out, int out_size, void* d_ws, size_t ws_size,
                              hipStream_t stream) {
    const float* x    = (const float*)d_in[0];
    const int*   pad  = (const int*)  d_in[1];
    const float* Wqkv = (const float*)d_in[2];
    const float* bqkv = (const float*)d_in[3];
    const float* Wo   = (const float*)d_in[4];
    const float* bo   = (const float*)d_in[5];
    float* out = (float*)d_out;

    float* qkv  = (float*)d_ws;                       // [4096, 1536]
    float* attn = qkv + (size_t)BATCH * SLEN * QKVN;  // [4096, 512]

    const int Mrows = BATCH * SLEN;                   // 4096

    // 1) QKV projection
    {
        const int tiles = (Mrows / 16) * (QKVN / 64);
        gemm_wmma<<<(tiles + 7) / 8, 256, 0, stream>>>(x, Wqkv, bqkv, qkv,
                                                       Mrows, EMB, QKVN);
    }
    // 2) Banded attention
    {
        const int waves = BATCH * NH * (SLEN / 16);   // 2048
        attn_banded<<<waves / 4, 128, 0, stream>>>(qkv, pad, attn);
    }
    // 3) Output projection
    {
        const int tiles = (Mrows / 16) * (EMB / 64);
        gemm_wmma<<<(tiles + 7) / 8, 256, 0, stream>>>(attn, Wo, bo, out,
                                                       Mrows, EMB, EMB);
    }
}